// WaveletTarget_47656957116913
// MI455X (gfx1250) — compile-verified
//
#include <hip/hip_runtime.h>

typedef __attribute__((ext_vector_type(2))) float v2f;
typedef __attribute__((ext_vector_type(8))) float v8f;

// db4 analysis filters (pywt 'periodization' convention):
//   c[k] = sum_j f[j] * x[(2k+1-j) mod N]
constexpr float CLO[8] = {
    -0.010597401784997278f,  0.032883011666982945f,  0.030841381835986965f,
    -0.18703481171888114f,  -0.02798376941698385f,   0.6308807679295904f,
     0.7148465705525415f,    0.23037781330885523f };
// DEC_HI[j] = (-1)^(j+1) * DEC_LO[7-j]
constexpr float CHI[8] = {
    -0.23037781330885523f,   0.7148465705525415f,   -0.6308807679295904f,
    -0.02798376941698385f,   0.18703481171888114f,   0.030841381835986965f,
    -0.032883011666982945f, -0.010597401784997278f };

// Precomputed per-lane A-matrix fragments for V_WMMA_F32_16X16X4_F32.
// A is the 16x40 weight band W[m][n] = f[2m+7-n] for n in [2m, 2m+7], else 0
// (input rows staged from base = 2*k0 - 6). 16x4 f32 A layout (wave32):
// lanes 0-15 hold (K=4i+0, K=4i+1) for M=lane; lanes 16-31 hold (K=4i+2, 4i+3).
struct alignas(16) FragTab { float v[10][32][2]; };

constexpr FragTab make_frag_tab(const float (&F)[8]) {
    FragTab t = {};
    for (int i = 0; i < 10; ++i)
        for (int lane = 0; lane < 32; ++lane) {
            const int m    = lane & 15;
            const int koff = (lane & 16) ? 2 : 0;
            const int n0   = 4 * i + koff;
            const int j0   = 2 * m + 7 - n0;
            const int j1   = j0 - 1;
            t.v[i][lane][0] = (j0 >= 0 && j0 < 8) ? F[j0] : 0.0f;
            t.v[i][lane][1] = (j1 >= 0 && j1 < 8) ? F[j1] : 0.0f;
        }
    return t;
}

__device__ constexpr FragTab TLO = make_frag_tab(CLO);
__device__ constexpr FragTab THI = make_frag_tab(CHI);

// One DWT level via V_WMMA_F32_16X16X4_F32.
// Grid: x = Na/32 (output-row tiles of 16), y = batch. Block: 128 (4 waves).
// Wave w computes 16 output rows x features [16w,16w+16) for both cA and cD:
// 10 K-steps of 4 over the 38-row band (padded to 40), one B fragment feeding
// two accumulators (low-pass -> cA, high-pass -> cD). 20 WMMAs per wave.
__global__ __launch_bounds__(128)
void dwt_level_wmma(const float* __restrict__ in, long in_bstride,
                    float* __restrict__ outA, long outA_bstride,
                    float* __restrict__ outD, long outD_bstride,
                    int Na)
{
    __shared__ float smem[40 * 72];   // [row 0..39][feat 0..63 + 8 pad], 11.25 KB
                                      // 72-word stride: float4-aligned rows and
                                      // conflict-free half-wave B-fragment reads

    const int k0   = blockIdx.x * 16;       // first output row of this tile
    const int b    = blockIdx.y;
    const int mask = Na - 1;                // Na is a power of two
    const int base = 2 * k0 - 6;            // first input row of the band

    const int lane = threadIdx.x & 31;
    const int wave = threadIdx.x >> 5;

    // ---- A fragments: branch-free coalesced b64 loads from constant tables ----
    v2f alo[10], ahi[10];
#pragma unroll
    for (int i = 0; i < 10; ++i) {
        alo[i] = *(const v2f*)&TLO.v[i][lane][0];
        ahi[i] = *(const v2f*)&THI.v[i][lane][0];
    }

    // ---- stage 38 input rows x 64 features into LDS (rows 38,39 zeroed) ----
    const float* inb = in + (size_t)b * in_bstride;
    for (int idx = threadIdx.x; idx < 640; idx += 128) {   // 40 rows * 16 float4
        const int row = idx >> 4;
        const int c4  = (idx & 15) << 2;
        float4 v;
        if (row < 38) {
            const int grow = (base + row) & mask;          // periodic wrap
            v = *(const float4*)(inb + (size_t)grow * 64 + c4);
        } else {
            v = make_float4(0.f, 0.f, 0.f, 0.f);
        }
        *(float4*)(&smem[row * 72 + c4]) = v;
    }
    __syncthreads();

    // ---- per-lane B/C/D coordinates ----
    const int m    = lane & 15;                  // feature within wave tile (B/C/D cols)
    const int koff = (lane & 16) ? 2 : 0;        // K pair selector
    const int col  = wave * 16 + m;              // feature column

    // ---- accumulate: D = A x B + C over the 40-wide band, shared B fragment ----
    v8f accA = {};
    v8f accD = {};
#pragma unroll
    for (int i = 0; i < 10; ++i) {
        const int r0 = 4 * i + koff;
        v2f bf;
        bf.x = smem[r0 * 72 + col];
        bf.y = smem[(r0 + 1) * 72 + col];
        accA = __builtin_amdgcn_wmma_f32_16x16x4_f32(
            false, alo[i], false, bf, (short)0, accA, false, false);
        accD = __builtin_amdgcn_wmma_f32_16x16x4_f32(
            false, ahi[i], false, bf, (short)0, accD, false, false);
    }

    // ---- store: C/D layout -> VGPR r holds rows (r | r+8) split at lane 16 ----
    const int orow = k0 + ((lane & 16) ? 8 : 0);
    float* pA = outA + (size_t)b * outA_bstride + (size_t)orow * 64 + col;
    float* pD = outD + (size_t)b * outD_bstride + (size_t)orow * 64 + col;
#pragma unroll
    for (int r = 0; r < 8; ++r) {
        pA[(size_t)r * 64] = accA[r];
        pD[(size_t)r * 64] = accD[r];
    }
}

extern "C" void kernel_launch(void* const* d_in, const int* in_sizes, int n_in,
                              void* d_out, int out_size, void* d_ws, size_t ws_size,
                              hipStream_t stream) {
    (void)in_sizes; (void)n_in; (void)out_size; (void)ws_size;

    const int B = 64, N = 4096, F = 64;
    const float* x  = (const float*)d_in[0];
    float* out = (float*)d_out;

    // workspace: cA1 (B x 2048 x F) then cA2 (B x 1024 x F)
    float* cA1 = (float*)d_ws;
    float* cA2 = cA1 + (size_t)B * (N / 2) * F;

    const long obs = (long)N * F;  // d_out batch stride (4096*64)

    // Level 1: x(4096) -> cA1(ws), cD1 -> out rows [2048,4096)
    dwt_level_wmma<<<dim3(N / 32, B), 128, 0, stream>>>(
        x,   (long)N * F,
        cA1, (long)(N / 2) * F,
        out + (size_t)(N / 2) * F, obs,
        N);

    // Level 2: cA1(2048) -> cA2(ws), cD2 -> out rows [1024,2048)
    dwt_level_wmma<<<dim3(N / 64, B), 128, 0, stream>>>(
        cA1, (long)(N / 2) * F,
        cA2, (long)(N / 4) * F,
        out + (size_t)(N / 4) * F, obs,
        N / 2);

    // Level 3: cA2(1024) -> cA3 -> out rows [0,512), cD3 -> out rows [512,1024)
    dwt_level_wmma<<<dim3(N / 128, B), 128, 0, stream>>>(
        cA2, (long)(N / 4) * F,
        out, obs,
        out + (size_t)(N / 8) * F, obs,
        N / 4);
}